// ChannelSpatial_30537217475261
// MI455X (gfx1250) — compile-verified
//
#include <hip/hip_runtime.h>

typedef float v2f __attribute__((ext_vector_type(2)));
typedef float v8f __attribute__((ext_vector_type(8)));

#define B_    16
#define C_    256
#define H_    64
#define W_    64
#define PLANE 4096            // H*W
#define NTOT  (B_*C_*PLANE)   // 16,777,216

// float-offset layout inside d_ws
#define WS_PSUM   0                   // 4096  per-(b,c) partial sums
#define WS_PSQ    4096                // 4096  per-(b,c) partial sum-of-squares
#define WS_SCALE  8192                // 256   inv_std * gamma
#define WS_SHIFT  8448                // 256   beta - mean*scale
#define WS_WBN    8704                // 256   |gamma| / sum|gamma|
#define WS_POOLED 9216                // B*2*PLANE = 131072
#define WS_GATE   (9216 + 131072)     // B*PLANE   = 65536

// ---------------- Kernel 1: per-plane partial stats (deterministic) ----------------
__global__ void __launch_bounds__(256)
k_stats_partial(const float* __restrict__ x, float* __restrict__ ws) {
    const int bc  = blockIdx.x;                 // b*C + c
    const int tid = threadIdx.x;
    const float* p = x + (size_t)bc * PLANE;
    float s = 0.f, q = 0.f;
#pragma unroll
    for (int i = 0; i < PLANE / 256; ++i) {     // 16 coalesced loads
        float v = p[tid + i * 256];
        s += v;
        q += v * v;
    }
    __shared__ float ls[256], lq[256];
    ls[tid] = s; lq[tid] = q;
    __syncthreads();
    for (int off = 128; off > 0; off >>= 1) {
        if (tid < off) { ls[tid] += ls[tid + off]; lq[tid] += lq[tid + off]; }
        __syncthreads();
    }
    if (tid == 0) {
        ws[WS_PSUM + bc] = ls[0];
        ws[WS_PSQ  + bc] = lq[0];
    }
}

// ---------------- Kernel 2: finalize stats + wbn ----------------
__global__ void __launch_bounds__(256)
k_finalize(const float* __restrict__ gamma, const float* __restrict__ beta,
           float* __restrict__ ws) {
    const int c = threadIdx.x;                  // 256 threads == channels
    float s = 0.f, q = 0.f;
#pragma unroll
    for (int b = 0; b < B_; ++b) {              // fixed-order sum -> deterministic
        s += ws[WS_PSUM + b * C_ + c];
        q += ws[WS_PSQ  + b * C_ + c];
    }
    const float invN = 1.0f / (float)(B_ * PLANE);
    float mean = s * invN;
    float var  = q * invN - mean * mean;
    float inv  = rsqrtf(var + 1e-5f);
    float g    = gamma[c];
    float scale = inv * g;
    ws[WS_SCALE + c] = scale;
    ws[WS_SHIFT + c] = beta[c] - mean * scale;

    __shared__ float ab[256];
    __shared__ float tot;
    float ga = fabsf(g);
    ab[c] = ga;
    __syncthreads();
    for (int off = 128; off > 0; off >>= 1) {
        if (c < off) ab[c] += ab[c + off];
        __syncthreads();
    }
    if (c == 0) tot = ab[0];
    __syncthreads();
    ws[WS_WBN + c] = ga / tot;
}

// ---------------- Kernel 3: fused normalize + channel avg/max pooling ----------------
__global__ void __launch_bounds__(256)
k_pooled(const float* __restrict__ x, float* __restrict__ ws) {
    __shared__ float sScale[C_], sShift[C_];
    const int tid = threadIdx.x;
    sScale[tid] = ws[WS_SCALE + tid];
    sShift[tid] = ws[WS_SHIFT + tid];
    __syncthreads();

    const int b  = blockIdx.x >> 4;
    const int pb = blockIdx.x & 15;
    const int p  = pb * 256 + tid;
    const float* xb = x + (size_t)b * C_ * PLANE + p;

    float sum = 0.f, mx = -3.402823466e38f;
#pragma unroll 4
    for (int c = 0; c < C_; ++c) {
        if (c + 4 < C_)
            __builtin_prefetch(xb + (size_t)(c + 4) * PLANE, 0, 0); // global_prefetch_b8
        float xh = xb[(size_t)c * PLANE] * sScale[c] + sShift[c];
        sum += xh;
        mx = fmaxf(mx, xh);
    }
    float* pooled = ws + WS_POOLED;
    pooled[(b * 2 + 0) * PLANE + p] = sum * (1.0f / C_);
    pooled[(b * 2 + 1) * PLANE + p] = mx;
}

// ---------------- Kernel 4: 7x7 conv (2->1) via V_WMMA_F32_16X16X4_F32 + sigmoid ----------------
// One block per (batch, 16-row band). LDS tile: 2ch x 22rows x 70cols, zero padded.
// Each wave computes 16 output pixels per group as 25 chained 16x16x4 f32 WMMAs:
// A[m][k] = patch value for pixel m, tap k; B[k][n] = w[k] for all n (columns identical).
__global__ void __launch_bounds__(256)
k_conv_wmma(const float* __restrict__ conv_w, const float* __restrict__ conv_b,
            float* __restrict__ ws) {
    __shared__ float tile[2 * 22 * 70];   // 3080 floats
    __shared__ float wT[100];             // 98 taps + 2 zero pads
    __shared__ int   offT[100];           // tap -> tile offset

    const int tid = threadIdx.x;
    const int b   = blockIdx.x;
    const int by  = blockIdx.y;           // row band [by*16, by*16+16)
    const float* pooled = ws + WS_POOLED + (size_t)b * 2 * PLANE;

    for (int i = tid; i < 2 * 22 * 70; i += 256) {
        int c  = i / 1540;
        int rm = i - c * 1540;
        int tr = rm / 70;
        int tc = rm - tr * 70;
        int gr = by * 16 - 3 + tr;
        int gc = tc - 3;
        float v = 0.f;
        if (gr >= 0 && gr < H_ && gc >= 0 && gc < W_)
            v = pooled[c * PLANE + gr * W_ + gc];
        tile[i] = v;
    }
    if (tid < 100) {
        float wv = 0.f;
        int   off = 0;
        if (tid < 98) {
            wv = conv_w[tid];
            int c  = tid / 49;
            int rm = tid - c * 49;
            int rr = rm / 7;
            int s  = rm - rr * 7;
            off = c * 1540 + rr * 70 + s;
        }
        wT[tid]   = wv;
        offT[tid] = off;
    }
    __syncthreads();

    const float bias = conv_b[0];
    const int wave = tid >> 5;
    const int lane = tid & 31;
    const int m    = lane & 15;           // A-matrix row held by this lane
    const int kofs = (lane >> 4) << 1;    // lanes 16-31 hold K+2,K+3 (ISA 7.12.2)
    float* gate = ws + WS_GATE + (size_t)b * PLANE;

    for (int g = wave; g < 64; g += 8) {  // 64 groups of 16 pixels per band
        int yloc  = g >> 2;
        int x0g   = (g & 3) << 4;
        int pbase = yloc * 70 + x0g + m;
        v8f acc = {0.f, 0.f, 0.f, 0.f, 0.f, 0.f, 0.f, 0.f};
#pragma unroll
        for (int ks = 0; ks < 25; ++ks) {
            int kA = ks * 4 + kofs;
            v2f a, bm;
            a.x  = tile[pbase + offT[kA]];
            a.y  = tile[pbase + offT[kA + 1]];
            bm.x = wT[kA];
            bm.y = wT[kA + 1];
            acc = __builtin_amdgcn_wmma_f32_16x16x4_f32(
                false, a, false, bm, (short)0, acc, false, false);
        }
        // Every D column is identical; lanes with N==0 (lane 0 -> rows 0..7,
        // lane 16 -> rows 8..15) extract the 16 results.
        if ((lane & 15) == 0) {
            int mb = (lane >> 4) * 8;
            int y  = by * 16 + yloc;
#pragma unroll
            for (int r = 0; r < 8; ++r) {
                float z = acc[r] + bias;
                gate[y * W_ + x0g + mb + r] = 1.0f / (1.0f + __expf(-z));
            }
        }
    }
}

// ---------------- Kernel 5: final elementwise (float4) ----------------
__global__ void __launch_bounds__(256)
k_final(const float* __restrict__ x, const float* __restrict__ ws,
        float* __restrict__ out) {
    const int gid = blockIdx.x * 256 + threadIdx.x;   // float4 index
    const size_t base = (size_t)gid * 4;
    const int plane = (int)(base >> 12);              // b*C + c
    const int c = plane & (C_ - 1);
    const int b = plane >> 8;
    const int p = (int)(base & (PLANE - 1));

    const float scale = ws[WS_SCALE + c];
    const float shift = ws[WS_SHIFT + c];
    const float wbn   = ws[WS_WBN + c];

    const float4 xv = *(const float4*)(x + base);
    const float4 gv = *(const float4*)(ws + WS_GATE + (size_t)b * PLANE + p);

    float4 ov;
    {   float xh = xv.x * scale + shift; float w = wbn + gv.x;
        ov.x = xv.x / (1.0f + __expf(-w * xh)); }
    {   float xh = xv.y * scale + shift; float w = wbn + gv.y;
        ov.y = xv.y / (1.0f + __expf(-w * xh)); }
    {   float xh = xv.z * scale + shift; float w = wbn + gv.z;
        ov.z = xv.z / (1.0f + __expf(-w * xh)); }
    {   float xh = xv.w * scale + shift; float w = wbn + gv.w;
        ov.w = xv.w / (1.0f + __expf(-w * xh)); }
    *(float4*)(out + base) = ov;
}

// ---------------- Launch ----------------
extern "C" void kernel_launch(void* const* d_in, const int* in_sizes, int n_in,
                              void* d_out, int out_size, void* d_ws, size_t ws_size,
                              hipStream_t stream) {
    const float* x      = (const float*)d_in[0];
    const float* gamma  = (const float*)d_in[1];
    const float* beta   = (const float*)d_in[2];
    const float* conv_w = (const float*)d_in[3];
    const float* conv_b = (const float*)d_in[4];
    float* out = (float*)d_out;
    float* ws  = (float*)d_ws;

    k_stats_partial<<<B_ * C_, 256, 0, stream>>>(x, ws);
    k_finalize<<<1, 256, 0, stream>>>(gamma, beta, ws);
    k_pooled<<<B_ * (PLANE / 256), 256, 0, stream>>>(x, ws);
    k_conv_wmma<<<dim3(B_, H_ / 16), 256, 0, stream>>>(conv_w, conv_b, ws);
    k_final<<<NTOT / 4 / 256, 256, 0, stream>>>(x, ws, out);
}